// BPRLoss_88081189307054
// MI455X (gfx1250) — compile-verified
//
#include <hip/hip_runtime.h>
#include <cstdint>

// Problem constants (B=8, L=513, V=32000 per the reference setup_inputs)
#define VCOLS   32000
#define NROWS   4096     // B*(L-1)
#define LFULL   513
#define THREADS 512

typedef float v2f __attribute__((ext_vector_type(2)));
typedef float v8f __attribute__((ext_vector_type(8)));

__device__ __forceinline__ float log_sigmoid_f(float x) {
    // stable: min(x,0) - log1p(exp(-|x|))
    float ax = fabsf(x);
    return fminf(x, 0.0f) - log1pf(expf(-ax));
}

// One workgroup per logits row i. Stages the full 125KB row + the 16KB target
// vector into LDS using gfx1250 async global->LDS B128 copies, then does the
// 4096 masked log_sigmoid accumulations from LDS, reduces with wave32 shuffles
// plus a V_WMMA_F32_16X16X4_F32 (ones-matrix column-sum, exact in f32), and
// writes one partial sum per row.
extern "C" __global__ void __launch_bounds__(THREADS)
bpr_row_kernel(const float* __restrict__ preds,
               const int*   __restrict__ tgt,
               float*       __restrict__ rowsum)
{
    extern __shared__ char smem[];
    float* lds_row = reinterpret_cast<float*>(smem);                       // 128000 B
    int*   lds_tgt = reinterpret_cast<int*>(smem + (size_t)VCOLS * 4);     // 16384 B
    float* lds_red = reinterpret_cast<float*>(smem + (size_t)VCOLS * 4
                                                   + (size_t)NROWS * 4);   // 64 B

    const int tid = threadIdx.x;
    const int row = blockIdx.x;            // 0..4095
    const int b   = row >> 9;              // row / 512
    const int l   = row & 511;             // row % 512  (skips l == 512)
    const float* rowPtr = preds + (size_t)(b * LFULL + l) * VCOLS;

    // ---- async stage: targets (4096 ints = 1024 x B128) ----
    {
        unsigned ldsT = (unsigned)(uintptr_t)lds_tgt;      // LDS byte address
        const int* tPtr = tgt;
        for (int k = tid; k < NROWS / 4; k += THREADS) {
            unsigned goff = (unsigned)(k * 16);
            unsigned loff = ldsT + (unsigned)(k * 16);
            asm volatile("global_load_async_to_lds_b128 %0, %1, %2 offset:0"
                         :: "v"(loff), "v"(goff), "s"(tPtr) : "memory");
        }
    }

    // ---- async stage: the logits row (8000 x B128 = 128000 B) ----
    {
        unsigned ldsR = (unsigned)(uintptr_t)lds_row;      // LDS byte address
        for (int k = tid; k < VCOLS / 4; k += THREADS) {
            unsigned goff = (unsigned)(k * 16);
            unsigned loff = ldsR + (unsigned)(k * 16);
            asm volatile("global_load_async_to_lds_b128 %0, %1, %2 offset:0"
                         :: "v"(loff), "v"(goff), "s"(rowPtr) : "memory");
        }
    }

    asm volatile("s_wait_asynccnt 0" ::: "memory");   // my wave's copies landed
    __syncthreads();                                  // everyone's copies landed

    // ---- compute: acc = sum_j mask_j * log_sigmoid(pos_i - row[tgt[j]]) ----
    const int ci  = lds_tgt[row];
    float acc = 0.0f;
    if (ci != 0) {                                    // row mask
        const float pos = lds_row[ci];
        for (int j = tid; j < NROWS; j += THREADS) {
            int c = lds_tgt[j];
            if (c != 0) {                             // column mask
                float d = pos - lds_row[c];
                acc += log_sigmoid_f(d);
            }
        }
    }

    // ---- per-wave reduction (wave32 shuffles), 16 partials -> LDS ----
    #pragma unroll
    for (int off = 16; off > 0; off >>= 1)
        acc += __shfl_down(acc, off, 32);
    if ((tid & 31) == 0) lds_red[tid >> 5] = acc;     // 16 waves
    __syncthreads();

    // ---- final 16-partial reduction via V_WMMA_F32_16X16X4_F32 ----
    // A = ones(16x4), B[k,n] = p[4n+k]  =>  D[0,n] = sum_k p[4n+k].
    // Executed by ALL waves (EXEC must be all 1s for WMMA); redundant but free.
    {
        const int lane = tid & 31;
        const int n    = lane & 15;
        const int k0   = (lane < 16) ? 0 : 2;         // lanes 16-31 supply K=2,3
        float p0 = 0.0f, p1 = 0.0f;
        if (n < 4) {
            p0 = lds_red[4 * n + k0];
            p1 = lds_red[4 * n + k0 + 1];
        }
        v2f a;  a[0] = 1.0f; a[1] = 1.0f;             // all-ones A (16x4)
        v2f bm; bm[0] = p0;  bm[1] = p1;              // B (4x16), cols 0..3 used
        v8f cacc = {};
        cacc = __builtin_amdgcn_wmma_f32_16x16x4_f32(
            /*neg_a=*/false, a, /*neg_b=*/false, bm,
            /*c_mod=*/(short)0, cacc, /*reuse_a=*/false, /*reuse_b=*/false);
        float v = cacc[0];                            // lanes 0..3: group sums q_n
        v += __shfl_down(v, 2, 32);
        v += __shfl_down(v, 1, 32);                   // lane 0: q0+q1+q2+q3
        if (tid == 0) rowsum[row] = v;
    }
}

// Deterministic final reduction: sum 4096 row partials in double, count K,
// emit -(sum) / max(K*K, 1) as f32.
extern "C" __global__ void __launch_bounds__(1024)
bpr_final_kernel(const float* __restrict__ rowsum,
                 const int*   __restrict__ tgt,
                 float*       __restrict__ out)
{
    __shared__ double sred[32];
    __shared__ int    cred[32];
    const int tid = threadIdx.x;

    double s = 0.0;
    int cnt = 0;
    for (int j = tid; j < NROWS; j += 1024) {
        s   += (double)rowsum[j];
        cnt += (tgt[j] != 0) ? 1 : 0;
    }
    #pragma unroll
    for (int off = 16; off > 0; off >>= 1) {
        s   += __shfl_down(s,   off, 32);
        cnt += __shfl_down(cnt, off, 32);
    }
    if ((tid & 31) == 0) { sred[tid >> 5] = s; cred[tid >> 5] = cnt; }
    __syncthreads();
    if (tid < 32) {
        double sv = sred[tid];
        int    cv = cred[tid];
        #pragma unroll
        for (int off = 16; off > 0; off >>= 1) {
            sv += __shfl_down(sv, off, 32);
            cv += __shfl_down(cv, off, 32);
        }
        if (tid == 0) {
            double K = (double)cv;
            double denom = K * K;
            if (denom < 1.0) denom = 1.0;
            out[0] = (float)(-sv / denom);
        }
    }
}

extern "C" void kernel_launch(void* const* d_in, const int* in_sizes, int n_in,
                              void* d_out, int out_size, void* d_ws, size_t ws_size,
                              hipStream_t stream)
{
    const float* preds = (const float*)d_in[0];   // [8, 513, 32000] f32
    const int*   tgt   = (const int*)d_in[1];     // [4096] int
    float* rowsum = (float*)d_ws;                 // 4096 f32 partials
    float* outp   = (float*)d_out;                // scalar f32

    const size_t shmem = (size_t)VCOLS * 4    // row buffer   128000 B
                       + (size_t)NROWS * 4    // target ints   16384 B
                       + 64;                  // reduction scratch

    bpr_row_kernel<<<NROWS, THREADS, shmem, stream>>>(preds, tgt, rowsum);
    bpr_final_kernel<<<1, 1024, 0, stream>>>(rowsum, tgt, outp);
}